// Gate_32203664785675
// MI455X (gfx1250) — compile-verified
//
#include <hip/hip_runtime.h>

#define DIM 2048
#define N_EXP 8
#define TOKENS 16384
#define TOPK 2
#define ALPHA 1e-4f
#define ROUTE_SCALE 1.0f

#define BLOCK_THREADS 256
#define TOK_PER_BLOCK 128           // 8 waves * 16 tokens
#define NBLOCKS (TOKENS / TOK_PER_BLOCK)   // 128

typedef __attribute__((ext_vector_type(2))) float v2f;
typedef __attribute__((ext_vector_type(8))) float v8f;

// ---------------------------------------------------------------------------
// Main kernel: scores via V_WMMA_F32_16X16X4_F32, softmax + top-2 + outputs,
// per-block partial reductions for the aux loss.
// ---------------------------------------------------------------------------
__global__ __launch_bounds__(BLOCK_THREADS)
void gate_main(const float* __restrict__ x,
               const float* __restrict__ weight,
               const float* __restrict__ bias,
               float* __restrict__ out_w,   // [TOKENS][TOPK]
               int*   __restrict__ out_i,   // [TOKENS][TOPK]
               float* __restrict__ ws_p,    // [NBLOCKS][N_EXP] partial prob sums
               int*   __restrict__ ws_c)    // [NBLOCKS][N_EXP] partial counts
{
    // experts 0..7 + one zero row (row 8) so padded N-lanes (8..15) read 0
    __shared__ float w_lds[(N_EXP + 1) * DIM];          // 72 KB
    __shared__ float s_val[TOK_PER_BLOCK][N_EXP];       // scores, then probs
    __shared__ int   s_top[TOK_PER_BLOCK][TOPK];

    const int tid = threadIdx.x;

    // Stage weights into LDS (coalesced), build the zero row.
    for (int i = tid; i < N_EXP * DIM; i += BLOCK_THREADS) w_lds[i] = weight[i];
    for (int i = tid; i < DIM; i += BLOCK_THREADS)         w_lds[N_EXP * DIM + i] = 0.0f;
    __syncthreads();

    const int wave = tid >> 5;
    const int lane = tid & 31;
    const int hi   = lane >> 4;      // 0: K pair (k,k+1)   1: K pair (k+2,k+3)
    const int nn   = lane & 15;      // A: token row in tile; B/C/D: expert col
    const int tok0 = blockIdx.x * TOK_PER_BLOCK + wave * 16;

    // A fragment source: row = tile token nn, columns advance with k
    const float* xrow = x + (size_t)(tok0 + nn) * DIM + hi * 2;
    // B fragment source: expert row nn (padded -> zero row), columns with k
    const int brow = (nn < N_EXP) ? nn : N_EXP;
    const float* wrow = &w_lds[brow * DIM + hi * 2];

    v8f c = {0.f, 0.f, 0.f, 0.f, 0.f, 0.f, 0.f, 0.f};

    #pragma unroll 8
    for (int k = 0; k < DIM; k += 4) {
        v2f a = __builtin_nontemporal_load((const v2f*)(xrow + k)); // x is stream-once
        v2f b = *(const v2f*)(wrow + k);                            // ds_load_b64
        // D = A(16x4) * B(4x16) + C   (exact f32, matches reference math)
        c = __builtin_amdgcn_wmma_f32_16x16x4_f32(
                /*neg_a=*/false, a, /*neg_b=*/false, b,
                /*c_mod=*/(short)0, c, /*reuse_a=*/false, /*reuse_b=*/false);
    }

    // C layout: VGPR j -> token row (hi*8 + j), column = nn (expert)
    if (nn < N_EXP) {
        #pragma unroll
        for (int j = 0; j < 8; ++j)
            s_val[wave * 16 + hi * 8 + j][nn] = c[j];
    }
    __syncthreads();

    // ---- per-token softmax / biased top-2 / gather (one thread per token) ----
    if (tid < TOK_PER_BLOCK) {
        float sc[N_EXP];
        #pragma unroll
        for (int e = 0; e < N_EXP; ++e) sc[e] = s_val[tid][e];

        float m = sc[0];
        #pragma unroll
        for (int e = 1; e < N_EXP; ++e) m = fmaxf(m, sc[e]);

        float p[N_EXP]; float sum = 0.f;
        #pragma unroll
        for (int e = 0; e < N_EXP; ++e) { p[e] = __expf(sc[e] - m); sum += p[e]; }
        const float inv = 1.0f / sum;
        #pragma unroll
        for (int e = 0; e < N_EXP; ++e) p[e] *= inv;

        float bb[N_EXP];
        #pragma unroll
        for (int e = 0; e < N_EXP; ++e) bb[e] = p[e] + bias[e];

        // top-1: strict > keeps lowest index on ties (matches jax top_k)
        int   i0 = 0; float v0 = bb[0]; float q0 = p[0];
        #pragma unroll
        for (int e = 1; e < N_EXP; ++e)
            if (bb[e] > v0) { v0 = bb[e]; i0 = e; q0 = p[e]; }
        // top-2 excluding i0
        int   i1 = -1; float v1 = -3.402823466e38f; float q1 = 0.f;
        #pragma unroll
        for (int e = 0; e < N_EXP; ++e)
            if (e != i0 && bb[e] > v1) { v1 = bb[e]; i1 = e; q1 = p[e]; }

        const int t = blockIdx.x * TOK_PER_BLOCK + tid;
        out_w[t * TOPK + 0] = q0 * ROUTE_SCALE;
        out_w[t * TOPK + 1] = q1 * ROUTE_SCALE;
        out_i[t * TOPK + 0] = i0;
        out_i[t * TOPK + 1] = i1;

        // stash probs + picks for the deterministic per-block reduction
        #pragma unroll
        for (int e = 0; e < N_EXP; ++e) s_val[tid][e] = p[e];
        s_top[tid][0] = i0;
        s_top[tid][1] = i1;
    }
    __syncthreads();

    // ---- per-block partials (thread e handles expert e; fixed order -> deterministic)
    if (tid < N_EXP) {
        float sp = 0.f; int cnt = 0;
        for (int t = 0; t < TOK_PER_BLOCK; ++t) {
            sp  += s_val[t][tid];
            cnt += (s_top[t][0] == tid) + (s_top[t][1] == tid);
        }
        ws_p[blockIdx.x * N_EXP + tid] = sp;
        ws_c[blockIdx.x * N_EXP + tid] = cnt;
    }
}

// ---------------------------------------------------------------------------
// Final fold: aux_loss = ALPHA * sum_e( mean_t(probs[t][e]) * N_EXP * cnt[e]/(T*K) )
// ---------------------------------------------------------------------------
__global__ void gate_aux(const float* __restrict__ ws_p,
                         const int*   __restrict__ ws_c,
                         float* __restrict__ aux_out)
{
    if (threadIdx.x == 0 && blockIdx.x == 0) {
        float aux = 0.f;
        for (int e = 0; e < N_EXP; ++e) {
            float sp = 0.f; int cnt = 0;
            for (int b = 0; b < NBLOCKS; ++b) {
                sp  += ws_p[b * N_EXP + e];
                cnt += ws_c[b * N_EXP + e];
            }
            const float Pi = sp / (float)TOKENS;
            const float ce = (float)cnt / (float)(TOKENS * TOPK);
            aux += Pi * (ce * (float)N_EXP);
        }
        aux_out[0] = aux * ALPHA;
    }
}

// ---------------------------------------------------------------------------
extern "C" void kernel_launch(void* const* d_in, const int* in_sizes, int n_in,
                              void* d_out, int out_size, void* d_ws, size_t ws_size,
                              hipStream_t stream)
{
    const float* x      = (const float*)d_in[0];   // [16384, 2048] f32
    const float* weight = (const float*)d_in[1];   // [8, 2048]     f32
    const float* bias   = (const float*)d_in[2];   // [8]           f32

    // d_out: [weights f32 16384*2][indices i32 16384*2][aux f32 1]
    float* out_w  = (float*)d_out;
    int*   out_i  = (int*)d_out + TOKENS * TOPK;
    float* out_ax = (float*)d_out + 2 * TOKENS * TOPK;

    float* ws_p = (float*)d_ws;                       // NBLOCKS*N_EXP floats
    int*   ws_c = (int*)d_ws + NBLOCKS * N_EXP;       // NBLOCKS*N_EXP ints

    gate_main<<<NBLOCKS, BLOCK_THREADS, 0, stream>>>(x, weight, bias,
                                                     out_w, out_i, ws_p, ws_c);
    gate_aux<<<1, 32, 0, stream>>>(ws_p, ws_c, out_ax);
}